// HeteroGNN_85194971283954
// MI455X (gfx1250) — compile-verified
//
#include <hip/hip_runtime.h>
#include <hip/hip_bf16.h>

// ---------------- problem constants (from reference) ----------------
#define NU  100000
#define NI  20000
#define NE  800000
#define UD  64
#define IDD 128
#define TD  16
#define HD  64
#define OD  64
#define HC  128   // H*HD = 2*64

typedef float v2f __attribute__((ext_vector_type(2)));
typedef float v8f __attribute__((ext_vector_type(8)));

// monotonic uint encoding of f32 for atomic max
__device__ __forceinline__ unsigned fenc(float f) {
    unsigned u = __float_as_uint(f);
    return (u & 0x80000000u) ? ~u : (u | 0x80000000u);
}
__device__ __forceinline__ float fdec(unsigned e) {
    return (e & 0x80000000u) ? __uint_as_float(e ^ 0x80000000u)
                             : __uint_as_float(~e);
}

// ---------------------------------------------------------------
// WMMA GEMM:  out[M,N] = A[M,K] @ W[K,N] + bias[N]
//
// V_WMMA_F32_16X16X4_F32; wave32. blockDim.x = (N/16)*32 so the
// block's waves tile all of N. Each wave owns TWO 16-row M tiles
// (m0, m0+16) -> two independent accumulator chains per wave and
// each B fragment feeds two WMMAs.
//
// B staged in LDS k-paired: sW2[(k/2)*N + n] = {W[k][n], W[k+1][n]}
// so a fragment load is ONE ds_load_b64 at linear index
// (k/2 + half)*N + n.
// ---------------------------------------------------------------
__global__ void gemm_wmma_bias(const float* __restrict__ A,
                               const float* __restrict__ W,
                               const float* __restrict__ bias,
                               float* __restrict__ out,
                               int M, int K, int N) {
    __shared__ v2f   sW2[64 * 128];     // (K/2)*N pairs, max 128/2 * 128
    __shared__ float sB[128];
    const int tid = threadIdx.x;
    const int KP = K >> 1;
    for (int i = tid; i < KP * N; i += blockDim.x) {
        const int kp = i / N, n = i - kp * N;
        v2f w;
        w.x = W[(2 * kp) * N + n];
        w.y = W[(2 * kp + 1) * N + n];
        sW2[i] = w;
    }
    for (int i = tid; i < N; i += blockDim.x) sB[i] = bias[i];
    __syncthreads();

    const int lane = tid & 31;
    const int wv   = tid >> 5;
    const int half = lane >> 4;      // 0 or 1
    const int lq   = lane & 15;
    const int n0   = wv * 16;

    int row0 = blockIdx.x * 32 + lq;        // tile 0 rows
    int row1 = row0 + 16;                   // tile 1 rows
    if (row0 >= M) row0 = M - 1;            // keep EXEC all-ones for WMMA
    if (row1 >= M) row1 = M - 1;
    const float* A0 = A + (size_t)row0 * K;
    const float* A1 = A + (size_t)row1 * K;

    v8f acc0 = {};
    v8f acc1 = {};
    const v2f* sWn = sW2 + half * N + n0 + lq;   // + (k/2)*N per step
    for (int k = 0; k < K; k += 4) {
        const int ka = k + 2 * half;
        const v2f a0 = *(const v2f*)(A0 + ka);   // 8B aligned (ka even)
        const v2f a1 = *(const v2f*)(A1 + ka);
        const v2f b  = sWn[(k >> 1) * N];        // one ds_load_b64
        acc0 = __builtin_amdgcn_wmma_f32_16x16x4_f32(
            false, a0, false, b, (short)0, acc0, false, false);
        acc1 = __builtin_amdgcn_wmma_f32_16x16x4_f32(
            false, a1, false, b, (short)0, acc1, false, false);
    }

    const float bv = sB[n0 + lq];
    const int mbase = blockIdx.x * 32;
#pragma unroll
    for (int r = 0; r < 8; ++r) {
        const int m = mbase + r + 8 * half;
        if (m < M) out[(size_t)m * N + n0 + lq] = acc0[r] + bv;
    }
#pragma unroll
    for (int r = 0; r < 8; ++r) {
        const int m = mbase + 16 + r + 8 * half;
        if (m < M) out[(size_t)m * N + n0 + lq] = acc1[r] + bv;
    }
}

// ---------------------------------------------------------------
// zero init (also serves as identity for encoded-max tables)
// ---------------------------------------------------------------
__global__ void zero_u32(unsigned* __restrict__ p, long long n) {
    long long i = (long long)blockIdx.x * blockDim.x + threadIdx.x;
    if (i < n) p[i] = 0u;
}

// ---------------------------------------------------------------
// Edge pass 1: per-edge logits + atomic segment max over dst.
// one edge per wave32; lane l owns channels j=4l..4l+3; head h = l>>4.
// xe computed on the fly: xe = ea[e,:16] @ we[16,128] (we in LDS).
// ---------------------------------------------------------------
__global__ void edge_logits(const float* __restrict__ xl,   // [Nsrc,128]
                            const float* __restrict__ xr,   // [Ndst,128]
                            const float* __restrict__ ea,   // [E,16]
                            const int*   __restrict__ src,
                            const int*   __restrict__ dst,
                            const float* __restrict__ we,   // [16,128]
                            const float* __restrict__ att,  // [128]
                            float* __restrict__ logits,     // [E,2]
                            unsigned* __restrict__ mtab,    // [Ndst,2] encoded
                            int nEdges) {
    __shared__ float sWe[16 * 128];
    __shared__ float sAtt[128];
    for (int i = threadIdx.x; i < 16 * 128; i += blockDim.x) sWe[i] = we[i];
    if (threadIdx.x < 128) sAtt[threadIdx.x] = att[threadIdx.x];
    __syncthreads();

    const int e = blockIdx.x * (blockDim.x >> 5) + (threadIdx.x >> 5);
    if (e >= nEdges) return;
    const int lane = threadIdx.x & 31;
    const int s = src[e];
    const int d = dst[e];
    const int j = lane * 4;

    const float4 xlv = *(const float4*)(xl + (size_t)s * HC + j);
    const float4 xrv = *(const float4*)(xr + (size_t)d * HC + j);

    // broadcast the 16 edge features across the wave
    float eav = (lane < TD) ? ea[(size_t)e * TD + lane] : 0.0f;
    float4 xe = make_float4(0.f, 0.f, 0.f, 0.f);
#pragma unroll
    for (int t = 0; t < TD; ++t) {
        const float ev = __shfl(eav, t, 32);
        const float4 wv = *(const float4*)(sWe + t * HC + j);
        xe.x += ev * wv.x; xe.y += ev * wv.y;
        xe.z += ev * wv.z; xe.w += ev * wv.w;
    }

    float4 z = make_float4(xlv.x + xrv.x + xe.x, xlv.y + xrv.y + xe.y,
                           xlv.z + xrv.z + xe.z, xlv.w + xrv.w + xe.w);
    z.x = z.x > 0.f ? z.x : 0.2f * z.x;
    z.y = z.y > 0.f ? z.y : 0.2f * z.y;
    z.z = z.z > 0.f ? z.z : 0.2f * z.z;
    z.w = z.w > 0.f ? z.w : 0.2f * z.w;

    const float4 av = *(const float4*)(sAtt + j);
    float p = z.x * av.x + z.y * av.y + z.z * av.z + z.w * av.w;
#pragma unroll
    for (int off = 1; off < 16; off <<= 1) p += __shfl_xor(p, off, 32);

    const int h = lane >> 4;
    if ((lane & 15) == 0) {
        logits[(size_t)e * 2 + h] = p;
        atomicMax(mtab + (size_t)d * 2 + h, fenc(p));
    }
}

// ---------------------------------------------------------------
// Edge pass 2: ex = exp(logit - max[dst]); atomic-add denom.
// one edge (both heads, float2) per thread; ex overwrites logits.
// ---------------------------------------------------------------
__global__ void edge_exp(float* __restrict__ logits,       // [E,2]
                         const int* __restrict__ dst,
                         const unsigned* __restrict__ mtab, // [Ndst,2] encoded
                         float* __restrict__ denom,         // [Ndst,2]
                         int nEdges) {
    const int e = blockIdx.x * blockDim.x + threadIdx.x;
    if (e >= nEdges) return;
    const int d = dst[e];
    const uint2 me = *(const uint2*)(mtab + (size_t)d * 2);
    float2 lg = *(const float2*)(logits + (size_t)e * 2);
    lg.x = __expf(lg.x - fdec(me.x));
    lg.y = __expf(lg.y - fdec(me.y));
    *(float2*)(logits + (size_t)e * 2) = lg;
    atomicAdd(denom + (size_t)d * 2 + 0, lg.x);
    atomicAdd(denom + (size_t)d * 2 + 1, lg.y);
}

// ---------------------------------------------------------------
// Edge pass 3: acc[dst] += xl[src] * alpha   (alpha = ex / denom[dst])
// one edge per wave; 4 channels per lane; f32 global atomics.
// ---------------------------------------------------------------
__global__ void edge_accum(const float* __restrict__ xl,
                           const float* __restrict__ ex,     // [E,2]
                           const float* __restrict__ denom,  // [Ndst,2]
                           const int* __restrict__ src,
                           const int* __restrict__ dst,
                           float* __restrict__ acc,          // [Ndst,128]
                           int nEdges) {
    const int e = blockIdx.x * (blockDim.x >> 5) + (threadIdx.x >> 5);
    if (e >= nEdges) return;
    const int lane = threadIdx.x & 31;
    const int h = lane >> 4;
    const int s = src[e];
    const int d = dst[e];
    const float alpha = ex[(size_t)e * 2 + h] /
                        (denom[(size_t)d * 2 + h] + 1e-16f);
    const int j = lane * 4;
    const float4 xlv = *(const float4*)(xl + (size_t)s * HC + j);
    float* o = acc + (size_t)d * HC + j;
    atomicAdd(o + 0, xlv.x * alpha);
    atomicAdd(o + 1, xlv.y * alpha);
    atomicAdd(o + 2, xlv.z * alpha);
    atomicAdd(o + 3, xlv.w * alpha);
}

// ---------------------------------------------------------------
// finalize: h = relu(acc + bias[128])
// ---------------------------------------------------------------
__global__ void relu_bias(float* __restrict__ acc,
                          const float* __restrict__ bias,
                          long long n) {
    const long long i = (long long)blockIdx.x * blockDim.x + threadIdx.x;
    if (i < n) {
        const float v = acc[i] + bias[i & (HC - 1)];
        acc[i] = v > 0.f ? v : 0.f;
    }
}

// ---------------------------------------------------------------
static inline void launch_gemm(const float* A, const float* W, const float* b,
                               float* out, int M, int K, int N,
                               hipStream_t s) {
    dim3 block((N / 16) * 32);
    dim3 grid((M + 31) / 32);
    gemm_wmma_bias<<<grid, block, 0, s>>>(A, W, b, out, M, K, N);
}

extern "C" void kernel_launch(void* const* d_in, const int* in_sizes, int n_in,
                              void* d_out, int out_size, void* d_ws, size_t ws_size,
                              hipStream_t stream) {
    const float* x_user    = (const float*)d_in[0];
    const float* x_item    = (const float*)d_in[1];
    const int*   eb_src    = (const int*)d_in[2];
    const int*   eb_dst    = (const int*)d_in[3];
    const int*   er_src    = (const int*)d_in[4];
    const int*   er_dst    = (const int*)d_in[5];
    const float* ea_buys   = (const float*)d_in[6];
    const float* ea_rev    = (const float*)d_in[7];
    const float* pu_w      = (const float*)d_in[8];
    const float* pu_b      = (const float*)d_in[9];
    const float* pi_w      = (const float*)d_in[10];
    const float* pi_b      = (const float*)d_in[11];
    const float* b_ll_w    = (const float*)d_in[12];
    const float* b_ll_b    = (const float*)d_in[13];
    const float* b_lr_w    = (const float*)d_in[14];
    const float* b_lr_b    = (const float*)d_in[15];
    const float* b_le_w    = (const float*)d_in[16];
    const float* b_att     = (const float*)d_in[17];
    const float* b_bias    = (const float*)d_in[18];
    const float* r_ll_w    = (const float*)d_in[19];
    const float* r_ll_b    = (const float*)d_in[20];
    const float* r_lr_w    = (const float*)d_in[21];
    const float* r_lr_b    = (const float*)d_in[22];
    const float* r_le_w    = (const float*)d_in[23];
    const float* r_att     = (const float*)d_in[24];
    const float* r_bias    = (const float*)d_in[25];
    const float* tu_w      = (const float*)d_in[26];
    const float* tu_b      = (const float*)d_in[27];
    const float* ti_w      = (const float*)d_in[28];
    const float* ti_b      = (const float*)d_in[29];

    float* ws = (float*)d_ws;
    size_t off = 0;
    float* hu   = ws + off; off += (size_t)NU * HD;     // [NU,64]
    float* hi   = ws + off; off += (size_t)NI * HD;     // [NI,64]
    float* xl_b = ws + off; off += (size_t)NU * HC;     // buys: src=users
    float* xr_b = ws + off; off += (size_t)NI * HC;     // buys: dst=items
    float* xl_r = ws + off; off += (size_t)NI * HC;     // rev:  src=items
    float* xr_r = ws + off; off += (size_t)NU * HC;     // rev:  dst=users
    float* lg_b = ws + off; off += (size_t)NE * 2;
    float* lg_r = ws + off; off += (size_t)NE * 2;
    // zero-initialized tail: [ m_b | dn_b | m_r | dn_r | acc_b | acc_r ]
    unsigned* m_b  = (unsigned*)(ws + off); off += (size_t)NI * 2;
    float*    dn_b = ws + off;              off += (size_t)NI * 2;
    unsigned* m_r  = (unsigned*)(ws + off); off += (size_t)NU * 2;
    float*    dn_r = ws + off;              off += (size_t)NU * 2;
    float*    acc_b = ws + off;             off += (size_t)NI * HC;  // item_h
    float*    acc_r = ws + off;             off += (size_t)NU * HC;  // user_h

    float* out_user = (float*)d_out;                   // [NU,64]
    float* out_item = (float*)d_out + (size_t)NU * OD; // [NI,64]

    // zero max/denom/accumulator region (encoded-max identity == 0)
    {
        const long long nz = (long long)(NI * 2 * 2 + NU * 2 * 2 +
                                         (size_t)NI * HC + (size_t)NU * HC);
        zero_u32<<<(unsigned)((nz + 255) / 256), 256, 0, stream>>>(m_b, nz);
    }

    // input projections
    launch_gemm(x_user, pu_w, pu_b, hu, NU, UD, HD, stream);
    launch_gemm(x_item, pi_w, pi_b, hi, NI, IDD, HD, stream);

    // per-relation linear transforms
    launch_gemm(hu, b_ll_w, b_ll_b, xl_b, NU, HD, HC, stream);
    launch_gemm(hi, b_lr_w, b_lr_b, xr_b, NI, HD, HC, stream);
    launch_gemm(hi, r_ll_w, r_ll_b, xl_r, NI, HD, HC, stream);
    launch_gemm(hu, r_lr_w, r_lr_b, xr_r, NU, HD, HC, stream);

    const int EB = 256;               // 8 edges / block
    const int egrid = (NE + 7) / 8;

    // relation 'buys' : users -> items
    edge_logits<<<egrid, EB, 0, stream>>>(xl_b, xr_b, ea_buys, eb_src, eb_dst,
                                          b_le_w, b_att, lg_b, m_b, NE);
    edge_exp<<<(NE + 255) / 256, 256, 0, stream>>>(lg_b, eb_dst, m_b, dn_b, NE);
    edge_accum<<<egrid, EB, 0, stream>>>(xl_b, lg_b, dn_b, eb_src, eb_dst, acc_b, NE);
    relu_bias<<<(unsigned)(((long long)NI * HC + 255) / 256), 256, 0, stream>>>(
        acc_b, b_bias, (long long)NI * HC);

    // relation 'rev_buys' : items -> users
    edge_logits<<<egrid, EB, 0, stream>>>(xl_r, xr_r, ea_rev, er_src, er_dst,
                                          r_le_w, r_att, lg_r, m_r, NE);
    edge_exp<<<(NE + 255) / 256, 256, 0, stream>>>(lg_r, er_dst, m_r, dn_r, NE);
    edge_accum<<<egrid, EB, 0, stream>>>(xl_r, lg_r, dn_r, er_src, er_dst, acc_r, NE);
    relu_bias<<<(unsigned)(((long long)NU * HC + 255) / 256), 256, 0, stream>>>(
        acc_r, r_bias, (long long)NU * HC);

    // output projections
    launch_gemm(acc_r, tu_w, tu_b, out_user, NU, HC, OD, stream);
    launch_gemm(acc_b, ti_w, ti_b, out_item, NI, HC, OD, stream);

    (void)in_sizes; (void)n_in; (void)out_size; (void)ws_size;
}